// NeuralHMM_67422396612846
// MI455X (gfx1250) — compile-verified
//
#include <hip/hip_runtime.h>
#include <hip/hip_bf16.h>
#include <math.h>

// ---------------------------------------------------------------------------
// Neural HMM forward pass for MI455X (gfx1250, wave32).
//   B=32, T=4096, K=16.  Outputs (concat): beliefs[B,T,K], log_beliefs[B,T,K],
//   log_normalizers[B,T], A[B,T,K,K].
// Strategy: chunked associative scan in linear space using
//   V_WMMA_F32_16X16X4_F32 for the 16x16x16 chunk matrix products.
// ---------------------------------------------------------------------------

typedef float v2f __attribute__((ext_vector_type(2)));
typedef float v4f __attribute__((ext_vector_type(4)));
typedef float v8f __attribute__((ext_vector_type(8)));

#define BATCH   32
#define TLEN    4096
#define KST     16
#define CHUNK   64
#define NCH     (TLEN / CHUNK)      // 64 chunks

#define ETA_MIN   0.02f
#define ETA_MAX   0.35f
#define RHO_C     0.001f
#define NEG_INF_C (-1000000.0f)

__device__ __forceinline__ float eta_of(float p) {
    float e = ETA_MIN + (ETA_MAX - ETA_MIN) * p;
    return fminf(fmaxf(e, 0.001f), 0.95f);
}

// ---------------------------------------------------------------------------
// Kernel 1: stream out the normalized tridiagonal transition matrices A.
// Pure bandwidth (134 MB). 128-bit nontemporal stores, fully coalesced.
// ---------------------------------------------------------------------------
__global__ void __launch_bounds__(256) build_A_kernel(
        const float* __restrict__ bp, float* __restrict__ Aout) {
    int idx4 = blockIdx.x * 256 + threadIdx.x;        // < 8,388,608
    int off4 = idx4 & 63;                             // 64 float4 per 16x16 tile
    int tile = idx4 >> 6;                             // b*T + t  (< 131072)
    float p   = bp[tile];
    float eta = eta_of(p);
    float stay_mid = fmaxf(1.0f - eta - RHO_C, 0.01f);
    float inv_mid  = 1.0f / fmaxf(stay_mid + eta + RHO_C, 1e-8f);
    int i  = off4 >> 2;            // row 0..15
    int j0 = (off4 & 3) << 2;      // col base 0,4,8,12
    v4f v;
#pragma unroll
    for (int q = 0; q < 4; ++q) {
        int j = j0 + q;
        float diag = (i == 0) ? (1.0f - eta)
                   : ((i == KST - 1) ? (1.0f - RHO_C) : stay_mid);
        float a = (j == i) ? diag
                : ((j == i + 1) ? eta
                : ((j == i - 1) ? RHO_C : 0.0f));
        float rsi = (i == 0 || i == KST - 1) ? 1.0f : inv_mid;
        v[q] = a * rsi;
    }
    __builtin_nontemporal_store(v, (v4f*)Aout + idx4);
}

// ---------------------------------------------------------------------------
// Kernel 2 (pass 1): per (batch, chunk) wave computes the chunk matrix product
//   Q_c = Prod_t N_t,  N_t[r][n] = clip(A_t[r][n]) * exp(em_t[n] - max em_t)
// Each 16x16x16 product = 4 x V_WMMA_F32_16X16X4_F32 (accumulating).
// Running product P (C/D layout) is re-shaped to the A-operand layout via LDS.
// ---------------------------------------------------------------------------
__global__ void __launch_bounds__(32) chunk_product_kernel(
        const float* __restrict__ em, const float* __restrict__ bp,
        float* __restrict__ Q) {
    __shared__ float lds[KST * KST];                  // 1 KB
    int b    = blockIdx.x >> 6;
    int c    = blockIdx.x & (NCH - 1);
    int lane = threadIdx.x;
    int half = lane >> 4;          // 0 or 1
    int m16  = lane & 15;

    // P in WMMA C/D layout: component v = row (v + 8*half), column m16.
    v8f P;
#pragma unroll
    for (int v = 0; v < 8; ++v) P[v] = ((v + 8 * half) == m16) ? 1.0f : 0.0f;

    const float* em_b = em + (long)b * TLEN * KST;
    const float* bp_b = bp + (long)b * TLEN;

    int t0 = c * CHUNK + (c == 0 ? 1 : 0);            // t=0 has no transition
    int t1 = (c + 1) * CHUNK;

    for (int t = t0; t < t1; ++t) {
        // ---- rescale P to keep the product in f32 range (scale cancels on
        //      the later normalize, so we don't need to track it) ----
        float mx = 1e-30f;
#pragma unroll
        for (int v = 0; v < 8; ++v) mx = fmaxf(mx, P[v]);
#pragma unroll
        for (int s = 16; s >= 1; s >>= 1) mx = fmaxf(mx, __shfl_xor(mx, s, 32));
        float inv = 1.0f / mx;

        // ---- P (C/D layout) -> row-major LDS ----
#pragma unroll
        for (int v = 0; v < 8; ++v) lds[(v + 8 * half) * KST + m16] = P[v] * inv;
        __syncthreads();

        // ---- load A-operand fragments of P:
        //      block kb covers columns [4kb..4kb+3]; this lane holds
        //      (row m16, cols 4kb + 2*half + {0,1})  => one b64 load each ----
        v2f afrag[4];
#pragma unroll
        for (int kb = 0; kb < 4; ++kb)
            afrag[kb] = *(const v2f*)&lds[m16 * KST + 4 * kb + 2 * half];
        __syncthreads();

        // ---- build N_t directly in B-operand layout from scalars ----
        float emv = em_b[t * KST + m16];
        float emmax = emv;
#pragma unroll
        for (int s = 16; s >= 1; s >>= 1) emmax = fmaxf(emmax, __shfl_xor(emmax, s, 32));
        float ep  = __expf(emv - emmax);              // column scale (states = cols)
        float eta = eta_of(bp_b[t]);
        float stay_mid = fmaxf(1.0f - eta - RHO_C, 0.01f);
        float inv_mid  = 1.0f / fmaxf(stay_mid + eta + RHO_C, 1e-8f);

        v8f acc = {};
#pragma unroll
        for (int kb = 0; kb < 4; ++kb) {
            v2f bfrag;
#pragma unroll
            for (int u = 0; u < 2; ++u) {
                int r = 4 * kb + u + 2 * half;        // B row within 4x16 block
                float diag = (r == 0) ? (1.0f - eta)
                           : ((r == KST - 1) ? (1.0f - RHO_C) : stay_mid);
                float a = (m16 == r) ? diag
                        : ((m16 == r + 1) ? eta
                        : ((m16 == r - 1) ? RHO_C : 0.0f));
                float rsi = (r == 0 || r == KST - 1) ? 1.0f : inv_mid;
                a = fmaxf(a * rsi, 1e-10f);           // matches log-clip of 1e-10
                bfrag[u] = a * ep;
            }
            // D = A(16x4) x B(4x16) + C   -- accumulate the 4 K-blocks
            acc = __builtin_amdgcn_wmma_f32_16x16x4_f32(
                      false, afrag[kb], false, bfrag, (short)0, acc, false, false);
        }
        P = acc;
    }

    // ---- write Q_c row-major to workspace ----
    float* q = Q + ((long)b * NCH + c) * (KST * KST);
#pragma unroll
    for (int v = 0; v < 8; ++v) q[(v + 8 * half) * KST + m16] = P[v];
}

// ---------------------------------------------------------------------------
// Kernel 3 (pass 2): per-batch sequential scan over the 64 chunk matrices.
// S[b][c] = normalized belief entering chunk c (chunk 0 starts one-hot).
// ---------------------------------------------------------------------------
__global__ void __launch_bounds__(32) chunk_scan_kernel(
        const float* __restrict__ Q, float* __restrict__ S) {
    int b    = blockIdx.x;
    int lane = threadIdx.x;
    int j    = lane & 15;
    float s  = (j == 0) ? 1.0f : 0.0f;                // belief at t=0 is one-hot
    float* Sb       = S + (long)b * NCH * KST;
    const float* Qb = Q + (long)b * NCH * (KST * KST);
    for (int c = 0; c < NCH; ++c) {
        if (lane < 16) Sb[c * KST + j] = s;
        // next start = normalize(s · Q_c)
        const float* q = Qb + c * (KST * KST);
        float pred = 0.0f;
#pragma unroll
        for (int i = 0; i < 16; ++i) {
            float si = __shfl(s, i, 32);
            pred += si * q[i * KST + j];
        }
        float tot = pred;                             // reduce within 16 states
#pragma unroll
        for (int m = 8; m >= 1; m >>= 1) tot += __shfl_xor(tot, m, 32);
        s = pred / fmaxf(tot, 1e-30f);
    }
}

// ---------------------------------------------------------------------------
// Kernel 4 (pass 3): per (batch, chunk) wave replays the chunk step-by-step
// using the tridiagonal structure and emits beliefs / log_beliefs / lZ.
// Lanes 16-31 mirror lanes 0-15 so neighbor shuffles are trivial.
// ---------------------------------------------------------------------------
__global__ void __launch_bounds__(32) emit_kernel(
        const float* __restrict__ em, const float* __restrict__ bp,
        const float* __restrict__ mk, const float* __restrict__ S,
        float* __restrict__ beliefs, float* __restrict__ logb,
        float* __restrict__ lZout) {
    int b    = blockIdx.x >> 6;
    int c    = blockIdx.x & (NCH - 1);
    int lane = threadIdx.x;
    int j    = lane & 15;
    float alpha = S[((long)b * NCH + c) * KST + j];

    const float* em_b = em + (long)b * TLEN * KST;
    const float* bp_b = bp + (long)b * TLEN;
    const float* mk_b = mk + (long)b * TLEN;
    float* bel_b = beliefs + (long)b * TLEN * KST;
    float* lb_b  = logb    + (long)b * TLEN * KST;
    float* lz_b  = lZout   + (long)b * TLEN;

    for (int tt = 0; tt < CHUNK; ++tt) {
        int t = c * CHUNK + tt;
        float emv = em_b[t * KST + j];
        float lj;
        if (t == 0) {
            lj = ((j == 0) ? 0.0f : NEG_INF_C) + emv; // exact t=0 semantics
        } else {
            float eta = eta_of(bp_b[t]);
            float stay_mid = fmaxf(1.0f - eta - RHO_C, 0.01f);
            float inv_mid  = 1.0f / fmaxf(stay_mid + eta + RHO_C, 1e-8f);
            float a_prev = __shfl(alpha, (lane - 1) & 31, 32);
            float a_next = __shfl(alpha, (lane + 1) & 31, 32);
            float ap = (j >= 1)  ? a_prev : 0.0f;
            float an = (j <= 14) ? a_next : 0.0f;
            // column j of A:  superdiag(row j-1)=eta, diag(row j), subdiag(row j+1)=rho
            float w_up = fmaxf(eta   * ((j - 1 == 0) ? 1.0f : inv_mid), 1e-10f);
            float diag = (j == 0) ? (1.0f - eta)
                       : ((j == 15) ? (1.0f - RHO_C) : stay_mid);
            float w_d  = fmaxf(diag  * ((j == 0 || j == 15) ? 1.0f : inv_mid), 1e-10f);
            float w_dn = fmaxf(RHO_C * ((j + 1 == 15) ? 1.0f : inv_mid), 1e-10f);
            // log-clip background: all other rows contribute 1e-10 * alpha_i
            float rest = fmaxf(1.0f - alpha - ap - an, 0.0f);
            float pred = ap * w_up + alpha * w_d + an * w_dn + 1e-10f * rest;
            lj = __logf(pred) + emv;
        }
        // logsumexp over the 16 states (halves are mirrored => 16-wide reduce)
        float mmax = lj;
#pragma unroll
        for (int m = 8; m >= 1; m >>= 1) mmax = fmaxf(mmax, __shfl_xor(mmax, m, 32));
        float z = __expf(lj - mmax);
#pragma unroll
        for (int m = 8; m >= 1; m >>= 1) z += __shfl_xor(z, m, 32);
        float lZ     = mmax + __logf(z);
        float lb_new = lj - lZ;
        float mval   = mk_b[t];
        float la_old = __logf(fmaxf(alpha, 1e-37f));
        float la     = mval * lb_new + (1.0f - mval) * la_old;
        alpha        = __expf(la);
        if (lane < 16) {
            bel_b[t * KST + j] = alpha;
            lb_b[t * KST + j]  = la;
        }
        if (lane == 0) lz_b[t] = mval * lZ;
    }
}

// ---------------------------------------------------------------------------
extern "C" void kernel_launch(void* const* d_in, const int* in_sizes, int n_in,
                              void* d_out, int out_size, void* d_ws, size_t ws_size,
                              hipStream_t stream) {
    const float* em = (const float*)d_in[0];   // emission_log_probs [B,T,K]
    const float* bp = (const float*)d_in[1];   // boundary_probs     [B,T]
    const float* mk = (const float*)d_in[2];   // mask               [B,T]

    float* out      = (float*)d_out;
    float* beliefs  = out;                                       // B*T*K
    float* logb     = out + (long)BATCH * TLEN * KST;            // B*T*K
    float* lZ       = logb + (long)BATCH * TLEN * KST;           // B*T
    float* Aout     = lZ + (long)BATCH * TLEN;                   // B*T*K*K

    // workspace: chunk products Q [B,NCH,16,16] then chunk-start beliefs S [B,NCH,16]
    float* Q = (float*)d_ws;                                     // 524288 floats
    float* S = Q + (long)BATCH * NCH * KST * KST;                // 32768 floats

    // 1) stream A: 8,388,608 x 128-bit NT stores
    build_A_kernel<<<32768, 256, 0, stream>>>(bp, Aout);
    // 2) chunk matrix products (WMMA), 2048 independent waves
    chunk_product_kernel<<<BATCH * NCH, 32, 0, stream>>>(em, bp, Q);
    // 3) per-batch scan over 64 chunk matrices
    chunk_scan_kernel<<<BATCH, 32, 0, stream>>>(Q, S);
    // 4) per-chunk replay + output emission, 2048 independent waves
    emit_kernel<<<BATCH * NCH, 32, 0, stream>>>(em, bp, mk, S, beliefs, logb, lZ);
}